// BilevelFramework_41669772706093
// MI455X (gfx1250) — compile-verified
//
#include <hip/hip_runtime.h>
#include <math.h>

#define N_NODES 110
#define NN      (N_NODES * N_NODES)   // 12100
#define NSEG    (NN + 1)              // 12101
#define TM_V4   (NN / 4)              // 3025 16-byte chunks (12100 % 4 == 0)
#define THREADS 256
#define PBLOCKS 512

// ---- order-preserving float<->uint map (atomicMax over floats of any sign) ----
__device__ __forceinline__ unsigned f2ord(float f) {
    int b = __float_as_int(f);
    return ((unsigned)b) ^ (((unsigned)(b >> 31)) | 0x80000000u);
}
__device__ __forceinline__ float ord2f(unsigned u) {
    unsigned b = (u & 0x80000000u) ? (u ^ 0x80000000u) : ~u;
    return __int_as_float((int)b);
}

// ---- stage time_mat (48.4 KB) into LDS via CDNA5 async global->LDS DMA ----
__device__ __forceinline__ void stage_time_mat(const float* __restrict__ tm,
                                               float* s_time) {
    // Generic LDS pointer: low 32 bits are the LDS byte offset (aperture rule).
    unsigned lds_base = (unsigned)(unsigned long long)(uintptr_t)s_time;
    for (int i = threadIdx.x; i < TM_V4; i += blockDim.x) {
        unsigned loff = lds_base + (unsigned)i * 16u;
        const float* g = tm + (size_t)i * 4;
        asm volatile("global_load_async_to_lds_b128 %0, %1, off"
                     :
                     : "v"(loff), "v"(g)
                     : "memory");
    }
    asm volatile("s_wait_asynccnt 0x0" ::: "memory");
    __syncthreads();
}

__device__ __forceinline__ float path_time(const float* s_time,
                                           const int* __restrict__ pu,
                                           const int* __restrict__ pv,
                                           const int* __restrict__ em,
                                           int p) {
    int base = p * 3;
    float t = 0.0f;
#pragma unroll
    for (int j = 0; j < 3; ++j) {
        int m = em[base + j];
        if (m) {
            int u = pu[base + j];
            int v = pv[base + j];
            t += s_time[u * N_NODES + v];   // ds_load_b32
        }
    }
    return t;
}

// ---- kernel 0: init workspace + output ----
__global__ __launch_bounds__(THREADS)
void k_init(unsigned* gmax_u, float* denom, float* out) {
    int i = blockIdx.x * blockDim.x + threadIdx.x;
    if (i < NSEG) {
        gmax_u[i] = 0u;      // ord-encoding minimum (acts as -inf sentinel)
        denom[i]  = 0.0f;
    }
    if (i < NN) out[i] = 0.0f;
}

// ================= Variant A: logit/e cached in workspace =================

__global__ __launch_bounds__(THREADS)
void k_max_cache(const float* __restrict__ tm,
                 const int* __restrict__ pu, const int* __restrict__ pv,
                 const int* __restrict__ em, const int* __restrict__ grp,
                 const int* __restrict__ lam_p,
                 unsigned* __restrict__ gmax_u,
                 float* __restrict__ cache, int P) {
    __shared__ __align__(16) float s_time[NN];
    stage_time_mat(tm, s_time);
    float lam = (float)lam_p[0];
    int stride = gridDim.x * blockDim.x;
    for (int p = blockIdx.x * blockDim.x + threadIdx.x; p < P; p += stride) {
        int g = grp[p];
        if (g == NN) continue;           // padding: avoid same-address atomic storm
        float logit = -lam * path_time(s_time, pu, pv, em, p);
        cache[p] = logit;
        atomicMax(&gmax_u[g], f2ord(logit));
    }
}

__global__ __launch_bounds__(THREADS)
void k_denom_cache(const int* __restrict__ grp,
                   const unsigned* __restrict__ gmax_u,
                   float* __restrict__ denom,
                   float* __restrict__ cache, int P) {
    int stride = gridDim.x * blockDim.x;
    for (int p = blockIdx.x * blockDim.x + threadIdx.x; p < P; p += stride) {
        int g = grp[p];
        if (g == NN) continue;
        float e = expf(cache[p] - ord2f(gmax_u[g]));
        cache[p] = e;                    // overwrite logit with exp term
        atomicAdd(&denom[g], e);
    }
}

__global__ __launch_bounds__(THREADS)
void k_flows_cache(const float* __restrict__ ode,
                   const int* __restrict__ pu, const int* __restrict__ pv,
                   const int* __restrict__ em,
                   const int* __restrict__ oo, const int* __restrict__ dd,
                   const int* __restrict__ grp,
                   const float* __restrict__ denom,
                   const float* __restrict__ cache,
                   float* __restrict__ out, int P) {
    int stride = gridDim.x * blockDim.x;
    for (int p = blockIdx.x * blockDim.x + threadIdx.x; p < P; p += stride) {
        int g = grp[p];
        if (g == NN) continue;
        int o = oo[p];
        int d = dd[p];
        float dem = ode[o * N_NODES + d];
        dem = (dem > 0.0f && o != d) ? dem : 0.0f;
        if (dem == 0.0f) continue;
        float den = denom[g];
        float c   = (den > 0.0f) ? dem * (cache[p] / den) : 0.0f;
        if (c == 0.0f) continue;
        int base = p * 3;
#pragma unroll
        for (int j = 0; j < 3; ++j) {
            if (em[base + j]) {
                int u = pu[base + j];
                int v = pv[base + j];
                atomicAdd(&out[u * N_NODES + v], c);  // global_atomic_add_f32
            }
        }
    }
}

// ============ Variant B: recompute (fallback if workspace too small) ============

__global__ __launch_bounds__(THREADS)
void k_max(const float* __restrict__ tm,
           const int* __restrict__ pu, const int* __restrict__ pv,
           const int* __restrict__ em, const int* __restrict__ grp,
           const int* __restrict__ lam_p,
           unsigned* __restrict__ gmax_u, int P) {
    __shared__ __align__(16) float s_time[NN];
    stage_time_mat(tm, s_time);
    float lam = (float)lam_p[0];
    int stride = gridDim.x * blockDim.x;
    for (int p = blockIdx.x * blockDim.x + threadIdx.x; p < P; p += stride) {
        int g = grp[p];
        if (g == NN) continue;
        float logit = -lam * path_time(s_time, pu, pv, em, p);
        atomicMax(&gmax_u[g], f2ord(logit));
    }
}

__global__ __launch_bounds__(THREADS)
void k_denom(const float* __restrict__ tm,
             const int* __restrict__ pu, const int* __restrict__ pv,
             const int* __restrict__ em, const int* __restrict__ grp,
             const int* __restrict__ lam_p,
             const unsigned* __restrict__ gmax_u,
             float* __restrict__ denom, int P) {
    __shared__ __align__(16) float s_time[NN];
    stage_time_mat(tm, s_time);
    float lam = (float)lam_p[0];
    int stride = gridDim.x * blockDim.x;
    for (int p = blockIdx.x * blockDim.x + threadIdx.x; p < P; p += stride) {
        int g = grp[p];
        if (g == NN) continue;
        float logit = -lam * path_time(s_time, pu, pv, em, p);
        atomicAdd(&denom[g], expf(logit - ord2f(gmax_u[g])));
    }
}

__global__ __launch_bounds__(THREADS)
void k_flows(const float* __restrict__ tm, const float* __restrict__ ode,
             const int* __restrict__ pu, const int* __restrict__ pv,
             const int* __restrict__ em,
             const int* __restrict__ oo, const int* __restrict__ dd,
             const int* __restrict__ grp, const int* __restrict__ lam_p,
             const unsigned* __restrict__ gmax_u,
             const float* __restrict__ denom,
             float* __restrict__ out, int P) {
    __shared__ __align__(16) float s_time[NN];
    stage_time_mat(tm, s_time);
    float lam = (float)lam_p[0];
    int stride = gridDim.x * blockDim.x;
    for (int p = blockIdx.x * blockDim.x + threadIdx.x; p < P; p += stride) {
        int g = grp[p];
        if (g == NN) continue;
        int o = oo[p];
        int d = dd[p];
        float dem = ode[o * N_NODES + d];
        dem = (dem > 0.0f && o != d) ? dem : 0.0f;
        if (dem == 0.0f) continue;
        float logit = -lam * path_time(s_time, pu, pv, em, p);
        float e   = expf(logit - ord2f(gmax_u[g]));
        float den = denom[g];
        float c   = (den > 0.0f) ? dem * (e / den) : 0.0f;
        if (c == 0.0f) continue;
        int base = p * 3;
#pragma unroll
        for (int j = 0; j < 3; ++j) {
            if (em[base + j]) {
                int u = pu[base + j];
                int v = pv[base + j];
                atomicAdd(&out[u * N_NODES + v], c);
            }
        }
    }
}

extern "C" void kernel_launch(void* const* d_in, const int* in_sizes, int n_in,
                              void* d_out, int out_size, void* d_ws, size_t ws_size,
                              hipStream_t stream) {
    const float* ode = (const float*)d_in[0];
    const float* tm  = (const float*)d_in[1];
    const int*   pu  = (const int*)d_in[2];
    const int*   pv  = (const int*)d_in[3];
    const int*   em  = (const int*)d_in[4];
    const int*   oo  = (const int*)d_in[5];
    const int*   dd  = (const int*)d_in[6];
    const int*   grp = (const int*)d_in[7];
    const int*   lam = (const int*)d_in[8];
    int P = in_sizes[2] / 3;

    unsigned* gmax_u = (unsigned*)d_ws;            // NSEG u32
    float*    denom  = (float*)(gmax_u + NSEG);    // NSEG f32
    float*    cache  = (float*)(gmax_u + 2 * NSEG);// P f32 (variant A only)
    float*    out    = (float*)d_out;

    k_init<<<(NSEG + THREADS - 1) / THREADS, THREADS, 0, stream>>>(gmax_u, denom, out);

    size_t need = (size_t)NSEG * 8 + (size_t)P * 4;
    if (ws_size >= need) {
        k_max_cache  <<<PBLOCKS, THREADS, 0, stream>>>(tm, pu, pv, em, grp, lam,
                                                       gmax_u, cache, P);
        k_denom_cache<<<PBLOCKS, THREADS, 0, stream>>>(grp, gmax_u, denom, cache, P);
        k_flows_cache<<<PBLOCKS, THREADS, 0, stream>>>(ode, pu, pv, em, oo, dd, grp,
                                                       denom, cache, out, P);
    } else {
        k_max  <<<PBLOCKS, THREADS, 0, stream>>>(tm, pu, pv, em, grp, lam, gmax_u, P);
        k_denom<<<PBLOCKS, THREADS, 0, stream>>>(tm, pu, pv, em, grp, lam, gmax_u,
                                                 denom, P);
        k_flows<<<PBLOCKS, THREADS, 0, stream>>>(tm, ode, pu, pv, em, oo, dd, grp, lam,
                                                 gmax_u, denom, out, P);
    }
}